// Scores_89395449299573
// MI455X (gfx1250) — compile-verified
//
#include <hip/hip_runtime.h>

// Problem constants (match reference)
#define N_ROWS 131072
#define DDIM   512
#define SDIM   28
#define KCHUNK 256   // K-values of W staged in LDS per phase (2 phases)
#define BLOCK  256   // 8 waves; each wave owns 32 rows -> 256 rows/block

typedef __attribute__((ext_vector_type(2))) float v2f;
typedef __attribute__((ext_vector_type(8))) float v8f;

// Fused kernel: scores = (thetas @ W) @ phi, out = (thetas @ W)[N-1, :]
// out layout: d_out[0..N-1] = scores, d_out[N..N+27] = last row of out_all.
__global__ __launch_bounds__(BLOCK) void scores_wmma_kernel(
    const float* __restrict__ thetas,
    const float* __restrict__ phi,
    const float* __restrict__ W,
    float* __restrict__ out)
{
  __shared__ float Wlds[KCHUNK * 32];  // 32 KB: W chunk, zero-padded to 32 cols
  __shared__ float phis[32];           // phi zero-padded to 32

  const int tid     = threadIdx.x;
  const int wave    = tid >> 5;
  const int lane    = tid & 31;
  const int lane_lo = lane & 15;
  const int half    = lane >> 4;       // 0: lanes 0-15, 1: lanes 16-31

  const int rowbase = blockIdx.x * (BLOCK / 32) * 32 + wave * 32;

  if (tid < 32) phis[tid] = (tid < SDIM) ? phi[tid] : 0.0f;

  // C/D fragments: acc[m][t] = rows [rowbase+16m, +16), cols [16t, 16t+16)
  v8f acc[2][2] = {};

  // A-fragment base: lane holds row (rowbase + 16m + lane_lo), K = {2*half, 2*half+1}
  const float* pA = thetas + (size_t)(rowbase + lane_lo) * DDIM + (half << 1);
  // B-fragment base offset inside LDS chunk: row K = 2*half (+1 in second vgpr), col n
  const unsigned wb = ((unsigned)half << 6) + (unsigned)lane_lo;  // (2*half)*32 + n

  for (int phase = 0; phase < DDIM / KCHUNK; ++phase) {
    __syncthreads();  // previous chunk fully consumed (and phis visible)
    const int k0 = phase * KCHUNK;
    // Cooperative, zero-padded load of W[k0..k0+KCHUNK) into LDS as [k][32]
    for (int i = tid; i < KCHUNK * 32; i += BLOCK) {
      const int kk = i >> 5;
      const int n  = i & 31;
      Wlds[i] = (n < SDIM) ? W[(size_t)(k0 + kk) * SDIM + n] : 0.0f;
    }
    __syncthreads();

#pragma unroll 4
    for (int kk = 0; kk < KCHUNK / 4; ++kk) {
      const int kg = k0 + (kk << 2);
      // A fragments (f32 16x4): one 8B load per lane per M-tile
      v2f a0 = *reinterpret_cast<const v2f*>(pA + kg);
      v2f a1 = *reinterpret_cast<const v2f*>(pA + 16 * DDIM + kg);
      // B fragments (f32 4x16) from LDS
      const unsigned idx = ((unsigned)kk << 7) + wb;  // (4*kk)*32 + wb
      v2f b0, b1;
      b0[0] = Wlds[idx];       b0[1] = Wlds[idx + 32];
      b1[0] = Wlds[idx + 16];  b1[1] = Wlds[idx + 48];

      acc[0][0] = __builtin_amdgcn_wmma_f32_16x16x4_f32(
          false, a0, false, b0, (short)0, acc[0][0], false, false);
      acc[0][1] = __builtin_amdgcn_wmma_f32_16x16x4_f32(
          false, a0, false, b1, (short)0, acc[0][1], false, false);
      acc[1][0] = __builtin_amdgcn_wmma_f32_16x16x4_f32(
          false, a1, false, b0, (short)0, acc[1][0], false, false);
      acc[1][1] = __builtin_amdgcn_wmma_f32_16x16x4_f32(
          false, a1, false, b1, (short)0, acc[1][1], false, false);
    }
  }

  // Epilogue: scores[row] = sum_n out_all[row][n] * phi[n]
  // C layout: vgpr r, lanes 0-15 -> (M=r, N=lane); lanes 16-31 -> (M=8+r, N=lane-16)
  const float p_lo = phis[lane_lo];
  const float p_hi = phis[lane_lo + 16];
#pragma unroll
  for (int m = 0; m < 2; ++m) {
#pragma unroll
    for (int r = 0; r < 8; ++r) {
      float p = acc[m][0][r] * p_lo + acc[m][1][r] * p_hi;
      // reduce across the 16 lanes of each half independently
      p += __shfl_xor(p, 1);
      p += __shfl_xor(p, 2);
      p += __shfl_xor(p, 4);
      p += __shfl_xor(p, 8);
      if (lane_lo == 0)
        out[rowbase + m * 16 + half * 8 + r] = p;
    }
  }

  // Last row of out_all: row N-1 = rowbase+31 -> m=1, M=15 -> vgpr 7, lanes 16-31
  if (rowbase + 32 == N_ROWS && half == 1) {
    out[N_ROWS + lane_lo] = acc[1][0][7];                       // cols 0..15
    if (lane_lo + 16 < SDIM) out[N_ROWS + 16 + lane_lo] = acc[1][1][7];  // cols 16..27
  }
}

extern "C" void kernel_launch(void* const* d_in, const int* in_sizes, int n_in,
                              void* d_out, int out_size, void* d_ws, size_t ws_size,
                              hipStream_t stream) {
  (void)in_sizes; (void)n_in; (void)out_size; (void)d_ws; (void)ws_size;
  const float* thetas = (const float*)d_in[0];
  const float* phi    = (const float*)d_in[1];
  const float* W      = (const float*)d_in[2];
  float* out = (float*)d_out;

  dim3 grid(N_ROWS / ((BLOCK / 32) * 32));  // 512 blocks
  scores_wmma_kernel<<<grid, BLOCK, 0, stream>>>(thetas, phi, W, out);
}